// FMSynth_1142461301240
// MI455X (gfx1250) — compile-verified
//
#include <hip/hip_runtime.h>
#include <math.h>

#define T_LEN   262144
#define BATCH   32
#define CHUNK   8192
#define NCHUNK  (T_LEN / CHUNK)     // 32 chunks per batch row
#define NFRAMES 250
#define TWO_PI_F 6.28318530717958647692f
#define K0      (TWO_PI_F / 16000.0f)   // 2*pi/SR
#define EPSV    1e-5f
#define FSTEP   (249.0f / 262143.0f)    // linspace(0,249,T) step

typedef float v2f __attribute__((ext_vector_type(2)));
typedef float v4f __attribute__((ext_vector_type(4)));
typedef float v8f __attribute__((ext_vector_type(8)));

__device__ __forceinline__ float clamp01(float x) {
    return fminf(fmaxf(x, 0.0f), 1.0f);
}

// Wave-uniform broadcast from a fixed lane: v_readlane_b32 -> SGPR.
// Much cheaper than __shfl (which lowers to ds_bpermute through the LDS pipe).
__device__ __forceinline__ float bcast_lane(float x, int srcLane) {
    return __uint_as_float(__builtin_amdgcn_readlane(__float_as_uint(x), srcLane));
}

// ---------------------------------------------------------------------------
// Pass 1: per-chunk partial sums of 2*pi*f0/SR.  grid = (NCHUNK, BATCH)
// ---------------------------------------------------------------------------
__global__ __launch_bounds__(256) void fm_pass1(const float* __restrict__ f0,
                                                float* __restrict__ chunkSum) {
    __shared__ float red[8];
    const int c = blockIdx.x, b = blockIdx.y;
    const int tid = threadIdx.x, lane = tid & 31, wid = tid >> 5;
    const v4f* p = (const v4f*)(f0 + (size_t)b * T_LEN + (size_t)c * CHUNK);
    float s = 0.0f;
#pragma unroll
    for (int i = 0; i < 8; ++i) {
        v4f x = p[i * 256 + tid];
        s += (x.x + x.y) + (x.z + x.w);
    }
#pragma unroll
    for (int o = 16; o > 0; o >>= 1) s += __shfl_xor(s, o, 32);
    if (lane == 0) red[wid] = s;
    __syncthreads();
    if (tid == 0) {
        float t = 0.0f;
#pragma unroll
        for (int w = 0; w < 8; ++w) t += red[w];
        chunkSum[b * NCHUNK + c] = K0 * t;
    }
}

// ---------------------------------------------------------------------------
// Pass 2: exclusive scan of chunk sums per row + ADSR envelope frame grids.
// grid = (BATCH)
// ---------------------------------------------------------------------------
__global__ __launch_bounds__(256) void fm_pass2(const float* __restrict__ adsr,
                                                const float* __restrict__ chunkSum,
                                                float* __restrict__ chunkOff,
                                                float* __restrict__ envF) {
    const int b = blockIdx.x, tid = threadIdx.x;
    if (tid == 0) {
        float run = 0.0f;
        for (int c = 0; c < NCHUNK; ++c) {
            chunkOff[b * NCHUNK + c] = run;
            run += chunkSum[b * NCHUNK + c];
        }
    }
    const float* ap = adsr + b * 11;
    for (int idx = tid; idx < 6 * NFRAMES; idx += 256) {
        const int k = idx / NFRAMES;
        const int f = idx - k * NFRAMES;
        const float fl = ap[k + 0], pk = ap[k + 1];
        const float a  = fmaxf(ap[k + 2], EPSV);
        const float d  = fmaxf(ap[k + 3], EPSV);
        const float su = ap[k + 4];
        const float r  = fmaxf(ap[k + 5], EPSV);
        const float t  = (float)f * (1.0f / 249.0f);
        const float att = clamp01(t / a);
        const float dec = clamp01((t - a) / d);
        const float rel = clamp01((t - (1.0f - r)) / r);
        const float env = att * (1.0f - (1.0f - su) * dec) * (1.0f - rel);
        envF[(b * 6 + k) * NFRAMES + f] = fl + (pk - fl) * env;
    }
}

// ---------------------------------------------------------------------------
// Pass 3: WMMA-based intra-chunk scan + FM operator chain.
// grid = (NCHUNK, BATCH), block = 256 (8 waves; wave handles 1024 samples as
// four 256-sample tiles; each tile is a 16x16 f32 matrix scanned via
// P = X * U with V_WMMA_F32_16X16X4_F32, U = upper-triangular ones).
// ---------------------------------------------------------------------------
__global__ __launch_bounds__(256) void fm_pass3(const float* __restrict__ mod_index,
                                                const float* __restrict__ freq_ratio,
                                                const float* __restrict__ f0,
                                                const float* __restrict__ chunkOff,
                                                const float* __restrict__ envF,
                                                float* __restrict__ out) {
    __shared__ float envS[6 * NFRAMES];
    __shared__ float waveSum[8];

    const int c = blockIdx.x, b = blockIdx.y;
    const int tid = threadIdx.x, lane = tid & 31, wid = tid >> 5;
    const int nIdx = lane & 15;        // N index (C/D layout), M index (A layout)
    const int half = lane >> 4;        // lane half selects K sub-pair

    // envelope frames -> LDS
    for (int idx = tid; idx < 6 * NFRAMES; idx += 256)
        envS[idx] = envF[b * 6 * NFRAMES + idx];

    // per-batch scalars
    float fr[6], mi[6];
#pragma unroll
    for (int k = 0; k < 6; ++k) {
        fr[k] = freq_ratio[b * 6 + k];
        mi[k] = 2.0f / (1.0f + __expf(-mod_index[b * 6 + k]));  // MAX_OL*sigmoid
    }

    // Constant triangular B slices: B[k][n] = (k <= n), K split into 4 slices.
    // 32-bit B 4x16 layout: lane n in [0,15]: v0=k0,v1=k1; lanes 16-31: v0=k2,v1=k3.
    v2f bm[4];
#pragma unroll
    for (int kk = 0; kk < 4; ++kk) {
        const int kg = 4 * kk + 2 * half;
        bm[kk].x = (kg     <= nIdx) ? 1.0f : 0.0f;
        bm[kk].y = (kg + 1 <= nIdx) ? 1.0f : 0.0f;
    }

    const float* xw = f0 + (size_t)b * T_LEN + (size_t)c * CHUNK + wid * 1024;

    v8f P[4];
    float tileOff = 0.0f;
#pragma unroll
    for (int tau = 0; tau < 4; ++tau) {
        const float* xt = xw + tau * 256;
        __builtin_prefetch(xt + 256, 0, 1);   // global_prefetch next tile
        v8f acc = {};
#pragma unroll
        for (int kk = 0; kk < 4; ++kk) {
            // A slice (16x4, f32): lane m = nIdx, lane half selects K pair.
            const int eoff = 16 * nIdx + 4 * kk + 2 * half;
            v2f a = *(const v2f*)(xt + eoff);
            a *= K0;
            acc = __builtin_amdgcn_wmma_f32_16x16x4_f32(
                false, a, false, bm[kk], (short)0, acc, false, false);
        }
        // row sums live in column N=15 (lanes 15 and 31); readlane -> SGPRs
        float r[16];
#pragma unroll
        for (int v = 0; v < 8; ++v) {
            r[v]     = bcast_lane(acc[v], 15);
            r[v + 8] = bcast_lane(acc[v], 31);
        }
        float offs[16];
        offs[0] = 0.0f;
#pragma unroll
        for (int i = 1; i < 16; ++i) offs[i] = offs[i - 1] + r[i - 1];
#pragma unroll
        for (int v = 0; v < 8; ++v)
            P[tau][v] = acc[v] + (half ? offs[v + 8] : offs[v]) + tileOff;
        tileOff += offs[15] + r[15];
    }

    if (lane == 0) waveSum[wid] = tileOff;   // wave total (uniform across lanes)
    __syncthreads();

    float base = chunkOff[b * NCHUNK + c];
#pragma unroll
    for (int w = 0; w < 8; ++w)
        if (w < wid) base += waveSum[w];

    float* op = out + (size_t)b * T_LEN + (size_t)c * CHUNK + wid * 1024;
    const int chunkT0 = c * CHUNK + wid * 1024;

#pragma unroll
    for (int tau = 0; tau < 4; ++tau) {
#pragma unroll
        for (int v = 0; v < 8; ++v) {
            const int e = tau * 256 + 16 * (v + 8 * half) + nIdx;
            const float om = P[tau][v] + base;          // inclusive cumsum phase

            // envelope resample (linear interp on 250-frame grid)
            const float pos = (float)(chunkT0 + e) * FSTEP;
            const float fi = floorf(pos);
            int i0 = (int)fi;
            if (i0 > 249) i0 = 249;
            int i1 = i0 + 1;
            if (i1 > 249) i1 = 249;
            const float frac = pos - fi;
            const float w1 = 1.0f - frac;
            float env[6];
#pragma unroll
            for (int k = 0; k < 6; ++k)
                env[k] = envS[k * NFRAMES + i0] * w1 + envS[k * NFRAMES + i1] * frac;

            // FM operator chain: 6 -> 5 -> 4 -> 3 ; 2 -> 1 ; out = (op3+op1)/2
            const float op6 = mi[5] * __sinf(fr[5] * om) * env[5];
            const float op5 = mi[4] * __sinf(fr[4] * om + TWO_PI_F * op6) * env[4];
            const float op4 = mi[3] * __sinf(fr[3] * om + TWO_PI_F * op5) * env[3];
            const float op3 = mi[2] * __sinf(fr[2] * om + TWO_PI_F * op4) * env[2];
            const float op2 = mi[1] * __sinf(fr[1] * om) * env[1];
            const float op1 = mi[0] * __sinf(fr[0] * om + TWO_PI_F * op2) * env[0];

            op[e] = (op3 + op1) * 0.5f;
        }
    }
}

// ---------------------------------------------------------------------------
// Workspace layout (floats):
//   [0,      1024)  chunkSum
//   [1024,   2048)  chunkOff
//   [2048,  50048)  envFrames [32][6][250]
// total 200192 bytes
// ---------------------------------------------------------------------------
extern "C" void kernel_launch(void* const* d_in, const int* in_sizes, int n_in,
                              void* d_out, int out_size, void* d_ws, size_t ws_size,
                              hipStream_t stream) {
    (void)in_sizes; (void)n_in; (void)out_size; (void)ws_size;
    const float* mod_index  = (const float*)d_in[0];
    const float* freq_ratio = (const float*)d_in[1];
    const float* f0         = (const float*)d_in[2];
    const float* adsr       = (const float*)d_in[3];
    float* out = (float*)d_out;
    float* ws  = (float*)d_ws;
    float* chunkSum = ws;
    float* chunkOff = ws + 1024;
    float* envF     = ws + 2048;

    fm_pass1<<<dim3(NCHUNK, BATCH), 256, 0, stream>>>(f0, chunkSum);
    fm_pass2<<<dim3(BATCH), 256, 0, stream>>>(adsr, chunkSum, chunkOff, envF);
    fm_pass3<<<dim3(NCHUNK, BATCH), 256, 0, stream>>>(mod_index, freq_ratio, f0,
                                                      chunkOff, envF, out);
}